// GCN_44564580663786
// MI455X (gfx1250) — compile-verified
//
#include <hip/hip_runtime.h>
#include <hip/hip_bf16.h>
#include <math.h>

// Problem constants (match reference)
#define F_IN 128
#define HDIM 64
#define NCLS 10
#define NGRAPH 256

typedef __attribute__((ext_vector_type(2))) float v2f;
typedef __attribute__((ext_vector_type(8))) float v8f;

// ---------------------------------------------------------------------------
// Init: deg = 1.0 (self-loop), pooled = 0, counts = 0
// ---------------------------------------------------------------------------
__global__ void k_init(float* __restrict__ deg, float* __restrict__ pooled,
                       float* __restrict__ counts, int N) {
    int i = blockIdx.x * blockDim.x + threadIdx.x;
    if (i < N) deg[i] = 1.0f;
    if (i < NGRAPH * HDIM) pooled[i] = 0.0f;
    if (i < NGRAPH) counts[i] = 0.0f;
}

// deg[dst[e]] += 1
__global__ void k_degree(const int* __restrict__ dst, float* __restrict__ deg, int E) {
    int e = blockIdx.x * blockDim.x + threadIdx.x;
    if (e < E) atomicAdd(&deg[dst[e]], 1.0f);
}

// dis = rsqrt(deg)
__global__ void k_rsqrt(const float* __restrict__ deg, float* __restrict__ dis, int N) {
    int i = blockIdx.x * blockDim.x + threadIdx.x;
    if (i < N) dis[i] = rsqrtf(deg[i]);
}

// norm_e[e] = dis[src]*dis[dst]
__global__ void k_edgenorm(const int* __restrict__ src, const int* __restrict__ dst,
                           const float* __restrict__ dis, float* __restrict__ norm_e, int E) {
    int e = blockIdx.x * blockDim.x + threadIdx.x;
    if (e < E) norm_e[e] = dis[src[e]] * dis[dst[e]];
}

// ---------------------------------------------------------------------------
// WMMA fp32 GEMM:  out[N x 64] = act(A[N x K]) @ W[K x 64]
// One wave computes a 16x16 tile using V_WMMA_F32_16X16X4_F32 over K/4 steps.
// RELU template fuses the previous layer's activation into the A load.
// Layouts per CDNA5 ISA 7.12.2 (f32, wave32):
//   A 16x4 : lanes 0-15 -> K=0,1 ; lanes 16-31 -> K=2,3  (v2f per lane)
//   B 4x16 : VGPR0 -> rows {0,2}, VGPR1 -> rows {1,3} striped over lane halves
//   C/D    : VGPR i -> row i (lanes 0-15) / row i+8 (lanes 16-31)
// ---------------------------------------------------------------------------
template <int K, bool RELU>
__global__ void k_gemm_wmma(const float* __restrict__ A, const float* __restrict__ W,
                            float* __restrict__ out, int nrows) {
    const int lane = threadIdx.x & 31;
    const int wave = blockIdx.x * (blockDim.x >> 5) + (threadIdx.x >> 5);
    const int mtile = wave >> 2;          // HDIM/16 == 4 N-tiles
    const int ntile = wave & 3;
    if (mtile * 16 >= nrows) return;      // whole wave exits together (EXEC all-1s kept)

    const int m0   = mtile * 16;
    const int n0   = ntile * 16;
    const int l15  = lane & 15;
    const int half = lane >> 4;           // 0 or 1
    const int row  = m0 + l15;            // A row for this lane
    const int col  = n0 + l15;            // B/D column for this lane

    v8f c = {};
    #pragma unroll 4
    for (int k = 0; k < K; k += 4) {
        const int ka = k + half * 2;      // this lane's K pair for A and B
        float a0 = A[row * K + ka];
        float a1 = A[row * K + ka + 1];
        if (RELU) { a0 = fmaxf(a0, 0.0f); a1 = fmaxf(a1, 0.0f); }
        v2f a; a.x = a0; a.y = a1;

        v2f b;
        b.x = W[(ka)     * HDIM + col];   // rows {0,2} of the 4-row B tile
        b.y = W[(ka + 1) * HDIM + col];   // rows {1,3}

        c = __builtin_amdgcn_wmma_f32_16x16x4_f32(
                /*neg_a=*/false, a, /*neg_b=*/false, b,
                /*c_mod=*/(short)0, c, /*reuse_a=*/false, /*reuse_b=*/false);
    }

    #pragma unroll
    for (int i = 0; i < 8; ++i) {
        const int r = m0 + i + half * 8;
        out[r * HDIM + col] = c[i];
    }
}

// h[n,f] = dis[n]^2 * hw[n,f] + b[f]   (self-loop term + bias, pre-scatter)
__global__ void k_selfbias(const float* __restrict__ hw, const float* __restrict__ dis,
                           const float* __restrict__ b, float* __restrict__ h, int N) {
    int idx = blockIdx.x * blockDim.x + threadIdx.x;
    if (idx >= N * HDIM) return;
    int n = idx >> 6, f = idx & 63;
    float d = dis[n];
    h[idx] = d * d * hw[idx] + b[f];
}

// h[dst[e], f..f+3] += norm_e[e] * hw[src[e], f..f+3]   (16 lanes / edge, float4)
__global__ void k_scatter(const float* __restrict__ hw, const int* __restrict__ src,
                          const int* __restrict__ dst, const float* __restrict__ norm_e,
                          float* __restrict__ h, int E) {
    int idx = blockIdx.x * blockDim.x + threadIdx.x;
    int e = idx >> 4;
    if (e >= E) return;
    int f = (idx & 15) << 2;
    float w = norm_e[e];
    const float4 v = *(const float4*)(hw + (size_t)src[e] * HDIM + f);
    float* o = h + (size_t)dst[e] * HDIM + f;
    atomicAdd(o + 0, v.x * w);
    atomicAdd(o + 1, v.y * w);
    atomicAdd(o + 2, v.z * w);
    atomicAdd(o + 3, v.w * w);
}

// counts[batch[n]] += 1
__global__ void k_counts(const int* __restrict__ batch, float* __restrict__ counts, int N) {
    int n = blockIdx.x * blockDim.x + threadIdx.x;
    if (n < N) atomicAdd(&counts[batch[n]], 1.0f);
}

// pooled[batch[n], f..f+3] += h[n, f..f+3]   (16 lanes / node, float4)
__global__ void k_pool(const float* __restrict__ h, const int* __restrict__ batch,
                       float* __restrict__ pooled, int N) {
    int idx = blockIdx.x * blockDim.x + threadIdx.x;
    int n = idx >> 4;
    if (n >= N) return;
    int f = (idx & 15) << 2;
    const float4 v = *(const float4*)(h + (size_t)n * HDIM + f);
    float* o = pooled + (size_t)batch[n] * HDIM + f;
    atomicAdd(o + 0, v.x);
    atomicAdd(o + 1, v.y);
    atomicAdd(o + 2, v.z);
    atomicAdd(o + 3, v.w);
}

// logits = (pooled/count) @ Wl + bl ; softmax  — one thread per graph (tiny)
__global__ void k_head(const float* __restrict__ pooled, const float* __restrict__ counts,
                       const float* __restrict__ Wl, const float* __restrict__ bl,
                       float* __restrict__ out) {
    int g = blockIdx.x * blockDim.x + threadIdx.x;
    if (g >= NGRAPH) return;
    float inv = 1.0f / fmaxf(counts[g], 1.0f);
    float logits[NCLS];
    #pragma unroll
    for (int c = 0; c < NCLS; ++c) logits[c] = bl[c];
    for (int f = 0; f < HDIM; ++f) {
        float p = pooled[g * HDIM + f] * inv;
        #pragma unroll
        for (int c = 0; c < NCLS; ++c) logits[c] = fmaf(p, Wl[f * NCLS + c], logits[c]);
    }
    float m = logits[0];
    #pragma unroll
    for (int c = 1; c < NCLS; ++c) m = fmaxf(m, logits[c]);
    float s = 0.0f, e[NCLS];
    #pragma unroll
    for (int c = 0; c < NCLS; ++c) { e[c] = expf(logits[c] - m); s += e[c]; }
    float is = 1.0f / s;
    #pragma unroll
    for (int c = 0; c < NCLS; ++c) out[g * NCLS + c] = e[c] * is;
}

// ---------------------------------------------------------------------------
extern "C" void kernel_launch(void* const* d_in, const int* in_sizes, int n_in,
                              void* d_out, int out_size, void* d_ws, size_t ws_size,
                              hipStream_t stream) {
    const float* x     = (const float*)d_in[0];
    const int*   edge  = (const int*)  d_in[1];   // [2, E]: row0=src, row1=dst
    const int*   batch = (const int*)  d_in[2];
    const float* W1 = (const float*)d_in[3];
    const float* b1 = (const float*)d_in[4];
    const float* W2 = (const float*)d_in[5];
    const float* b2 = (const float*)d_in[6];
    const float* W3 = (const float*)d_in[7];
    const float* b3 = (const float*)d_in[8];
    const float* Wl = (const float*)d_in[9];
    const float* bl = (const float*)d_in[10];
    float* out = (float*)d_out;

    const int N = in_sizes[0] / F_IN;     // 100000
    const int E = in_sizes[1] / 2;        // 1600000
    const int* src = edge;
    const int* dst = edge + E;

    // Workspace layout (all float, 16B aligned segments)
    float* deg    = (float*)d_ws;                 // N
    float* dis    = deg    + N;                   // N
    float* norm_e = dis    + N;                   // E
    float* hw     = norm_e + E;                   // N*HDIM  (h @ W scratch)
    float* h      = hw     + (size_t)N * HDIM;    // N*HDIM  (layer output)
    float* pooled = h      + (size_t)N * HDIM;    // NGRAPH*HDIM
    float* counts = pooled + NGRAPH * HDIM;       // NGRAPH

    const int T = 256;
    const int gN   = (N + T - 1) / T;
    const int gE   = (E + T - 1) / T;
    const int gNH  = (N * HDIM + T - 1) / T;
    const int gE16 = (E * 16 + T - 1) / T;        // scatter: 16 lanes/edge
    const int gN16 = (N * 16 + T - 1) / T;        // pool: 16 lanes/node
    const int gGem = ((N / 16) * (HDIM / 16) + (T / 32) - 1) / (T / 32); // 1 wave per 16x16 tile

    // --- normalization prep ---
    k_init    <<<gN, T, 0, stream>>>(deg, pooled, counts, N);
    k_degree  <<<gE, T, 0, stream>>>(dst, deg, E);
    k_rsqrt   <<<gN, T, 0, stream>>>(deg, dis, N);
    k_edgenorm<<<gE, T, 0, stream>>>(src, dst, dis, norm_e, E);

    // --- layer 1: x[N,128] @ W1 ---
    k_gemm_wmma<F_IN, false><<<gGem, T, 0, stream>>>(x, W1, hw, N);
    k_selfbias<<<gNH, T, 0, stream>>>(hw, dis, b1, h, N);
    k_scatter <<<gE16, T, 0, stream>>>(hw, src, dst, norm_e, h, E);

    // --- layer 2: relu(h) @ W2 ---
    k_gemm_wmma<HDIM, true><<<gGem, T, 0, stream>>>(h, W2, hw, N);
    k_selfbias<<<gNH, T, 0, stream>>>(hw, dis, b2, h, N);
    k_scatter <<<gE16, T, 0, stream>>>(hw, src, dst, norm_e, h, E);

    // --- layer 3: relu(h) @ W3 (no output relu) ---
    k_gemm_wmma<HDIM, true><<<gGem, T, 0, stream>>>(h, W3, hw, N);
    k_selfbias<<<gNH, T, 0, stream>>>(hw, dis, b3, h, N);
    k_scatter <<<gE16, T, 0, stream>>>(hw, src, dst, norm_e, h, E);

    // --- global mean pool + classifier head ---
    k_counts<<<gN, T, 0, stream>>>(batch, counts, N);
    k_pool  <<<gN16, T, 0, stream>>>(h, batch, pooled, N);
    k_head  <<<(NGRAPH + T - 1) / T, T, 0, stream>>>(pooled, counts, Wl, bl, out);
}